// GNNPolicy_71562745086094
// MI455X (gfx1250) — compile-verified
//
#include <hip/hip_runtime.h>
#include <hip/hip_bf16.h>

typedef float v2f __attribute__((ext_vector_type(2)));
typedef float v8f __attribute__((ext_vector_type(8)));

// The whole GNN collapses: LayerNorm over a size-1 feature axis outputs just
// its bias, so every constraint node embedding is the identical 16-vector
//   h2 = relu(W2 @ relu(ln_b * w1 + b1) + b2)
// and sort-pool/top-k of identical rows tiles h2 K times. Output is a single
// sigmoid scalar broadcast to all graphs. We do the 16x16 f32 layer with
// V_WMMA_F32_16X16X4_F32 (exact f32 WMMA path), one wave per block.
__global__ __launch_bounds__(32) void gnn_policy_collapsed(
    const float* __restrict__ cons_ln_b,
    const float* __restrict__ cons_w1,
    const float* __restrict__ cons_b1,
    const float* __restrict__ cons_w2,   // (16,16) row-major: w2[i*16+j]
    const float* __restrict__ cons_b2,
    const float* __restrict__ final_w,   // (1, 256)
    const float* __restrict__ final_b,
    float* __restrict__ out, int out_n)
{
    const int lane = threadIdx.x;      // 0..31 (wave32)
    const int half = lane >> 4;        // 0 => rows M=0..7 / K pair {0,1}; 1 => M=8..15 / K pair {2,3}
    const int m    = lane & 15;        // A-matrix row for this lane

    // ---- Layer 1 (uniform across lanes): h1[j] = relu(ln_b * w1[j] + b1[j]) ----
    const float bln = cons_ln_b[0];
    float h1[16];
#pragma unroll
    for (int j = 0; j < 16; ++j) {
        float v = fmaf(bln, cons_w1[j], cons_b1[j]);
        h1[j] = v > 0.0f ? v : 0.0f;
    }

    // ---- Layer 2: D = W2(16x16) x h1, via 4 chained V_WMMA_F32_16X16X4_F32 ----
    // C initialized with bias broadcast across all N columns: C[M][N] = b2[M].
    // C/D layout: VGPR r, lanes 0-15 hold M=r, lanes 16-31 hold M=r+8.
    v8f acc;
#pragma unroll
    for (int r = 0; r < 8; ++r) acc[r] = cons_b2[r + 8 * half];

#pragma unroll
    for (int kk = 0; kk < 4; ++kk) {
        // A 16x4 layout: lanes 0-15 -> {K0,K1}, lanes 16-31 -> {K2,K3}
        const int k0 = 4 * kk + (half ? 2 : 0);
        v2f a, b;
        a[0] = cons_w2[m * 16 + k0];
        a[1] = cons_w2[m * 16 + k0 + 1];
        // B 4x16: broadcast h1 across every N column so all D columns = W2@h1;
        // rows mirror A's K striping (lanes 0-15 -> {K0,K1}, 16-31 -> {K2,K3}).
        b[0] = h1[k0];
        b[1] = h1[k0 + 1];
        acc = __builtin_amdgcn_wmma_f32_16x16x4_f32(
            /*neg_a=*/false, a, /*neg_b=*/false, b,
            /*c_mod=*/(short)0, acc, /*reuse_a=*/false, /*reuse_b=*/false);
    }

    // ---- ReLU + sort-pool head ----
    // s = final_b + sum_i h2[i] * fw_sum[i], fw_sum[i] = sum_{k=0..15} final_w[k*16+i]
    float p = 0.0f;
#pragma unroll
    for (int r = 0; r < 8; ++r) {
        const int i = r + 8 * half;
        float fw = 0.0f;
#pragma unroll
        for (int k = 0; k < 16; ++k) fw += final_w[k * 16 + i];
        const float h2 = acc[r] > 0.0f ? acc[r] : 0.0f;
        p = fmaf(h2, fw, p);
    }
    // Each half-wave holds an identical partial over its 8 rows; add the other half.
    p += __shfl_xor(p, 16, 32);
    const float s = p + final_b[0];
    const float y = 1.0f / (1.0f + __expf(-s));

    const int idx = blockIdx.x * 32 + lane;
    if (idx < out_n) out[idx] = y;
}

extern "C" void kernel_launch(void* const* d_in, const int* in_sizes, int n_in,
                              void* d_out, int out_size, void* d_ws, size_t ws_size,
                              hipStream_t stream) {
    // setup_inputs() order:
    //  0 constraint_features, 1 edge_indices, 2 edge_features, 3 variable_features,
    //  4 batch, 5 ptr, 6 cons_ln_w, 7 cons_ln_b, 8 cons_w1, 9 cons_b1,
    // 10 cons_w2, 11 cons_b2, 12 edge_ln_w, 13 edge_ln_b, 14 var_ln_w,
    // 15 var_ln_b, 16 var_w1, 17 var_b1, 18 var_w2, 19 var_b2,
    // 20 final_w, 21 final_b
    const float* cons_ln_b = (const float*)d_in[7];
    const float* cons_w1   = (const float*)d_in[8];
    const float* cons_b1   = (const float*)d_in[9];
    const float* cons_w2   = (const float*)d_in[10];
    const float* cons_b2   = (const float*)d_in[11];
    const float* final_w   = (const float*)d_in[20];
    const float* final_b   = (const float*)d_in[21];
    float* out = (float*)d_out;

    const int blocks = (out_size + 31) / 32;  // 4096 outputs -> 128 single-wave blocks
    gnn_policy_collapsed<<<blocks, 32, 0, stream>>>(
        cons_ln_b, cons_w1, cons_b1, cons_w2, cons_b2, final_w, final_b,
        out, out_size);
}